// ChamferDistance_sumknn_42082089566514
// MI455X (gfx1250) — compile-verified
//
#include <hip/hip_runtime.h>

// ChamferDistance + sum-kNN for B=1, N=M=8192, D=3 on MI455X (gfx1250).
//
// Roofline: materializing the three 8192x8192 distance matrices would cost
// ~800MB of HBM traffic (~35us at 23.3 TB/s) for ~1.2 GFLOP of math, so all
// reductions (row-min, col-argmin, col-top4) are fused into the tile loops
// and nothing larger than O(N) touches memory.  Cross terms run on
// V_WMMA_F32_16X16X4_F32 with the spare K=3 slot carrying squared norms:
//   rowmin: A=(-2x0,-2x1,-2x2,1), B=(y0,y1,y2,ysq)  -> c = ysq - 2*x.y
//   colmin: A=(x0,x1,x2,xsq),     B=(-2y0,-2y1,-2y2,1) -> c = xsq - 2*x.y
//
// Points are pre-packed once as float4 (p0,p1,p2,psq) in global memory, and
// each block stages the packed 128KB array into LDS with one Tensor Data
// Mover transfer (tensor_load_to_lds + s_wait_tensorcnt).  The hot loop is:
//   ds_load_b64 (lane picks .xy or .zw) -> v_wmma -> 8x v_min/cndmask.

typedef __attribute__((ext_vector_type(2))) float v2f;
typedef __attribute__((ext_vector_type(4))) float v4f;
typedef __attribute__((ext_vector_type(8))) float v8f;
typedef __attribute__((ext_vector_type(4))) unsigned int u32x4;
typedef __attribute__((ext_vector_type(8))) unsigned int u32x8;

#define NPTS  8192
#define NTILE (NPTS / 16)            // 512 16-wide tiles per dimension
#define NWORDS (NPTS * 4)            // 32768 dwords = 128KB staged per block
#define STAGE_BYTES (NWORDS * 4)
#define FBIG  3.402823466e38f

extern __shared__ float g_lds[];     // TDM destination (dynamic LDS)

__device__ __forceinline__ v8f wmma_f32_16x16x4(v2f a, v2f b, v8f c) {
  // 8 args: (neg_a, A, neg_b, B, c_mod, C, reuse_a, reuse_b)
  return __builtin_amdgcn_wmma_f32_16x16x4_f32(false, a, false, b, (short)0, c,
                                               false, false);
}

// raw v_min_num_f32: skips the v_max canonicalize fminf() inserts (WMMA
// results are never sNaN, so minimumNumber semantics are unaffected).
__device__ __forceinline__ float fmin_fast(float a, float b) {
  float r;
  asm("v_min_num_f32 %0, %1, %2" : "=v"(r) : "v"(a), "v"(b));
  return r;
}

// one-instruction xor-lane exchange via ds_swizzle group-of-32 mode
template <int XM>
__device__ __forceinline__ float swz_xor_f(float v) {
  return __int_as_float(
      __builtin_amdgcn_ds_swizzle(__float_as_int(v), (XM << 10) | 0x1F));
}
template <int XM>
__device__ __forceinline__ int swz_xor_i(int v) {
  return __builtin_amdgcn_ds_swizzle(v, (XM << 10) | 0x1F);
}

// Stage NWORDS dwords from global into LDS with the Tensor Data Mover.
// D# group0: count=1, lds_addr, global_addr[56:0], type=2 ("image").
// D# group1: data_size=4B, tensor = one row of NWORDS dwords, tile = same.
__device__ __forceinline__ void tdm_stage(const float* __restrict__ src,
                                          float* dst_lds) {
  const unsigned long long ga = (unsigned long long)src;
  u32x4 g0;
  g0[0] = 1u;                                       // count=1, user mode
  g0[1] = (unsigned)(unsigned long long)dst_lds;    // lds_addr (bytes)
  g0[2] = (unsigned)ga;                             // global_addr[31:0]
  g0[3] = ((unsigned)(ga >> 32) & 0x01FFFFFFu)      // global_addr[56:32]
          | 0x80000000u;                            // type=2 at bits 127:126
  u32x8 g1;
  g1[0] = 0x00020000u;                 // data_size=2 (4B); no pad/iterate/mc
  g1[1] = (NWORDS & 0xFFFFu) << 16;    // tensor_dim0[15:0]   -> bits 63:48
  g1[2] = (NWORDS >> 16) | (1u << 16); // dim0[31:16], tensor_dim1=1
  g1[3] = (NWORDS & 0xFFFFu) << 16;    // tile_dim0 = NWORDS  -> bits 127:112
  g1[4] = 1u;                          // tile_dim1=1, tile_dim2=0
  g1[5] = NWORDS;                      // tensor_dim0_stride[31:0]
  g1[6] = 0u;
  g1[7] = 0u;
  asm volatile("tensor_load_to_lds %0, %1" :: "s"(g0), "s"(g1) : "memory");
  __builtin_amdgcn_s_wait_tensorcnt(0);
}

// value-then-lowest-index ordering (jax top_k / argmin tie-breaking) — only
// needed in cross-lane merges; in-lane streams have ascending indices.
__device__ __forceinline__ bool pless(float v, int i, float w, int j) {
  return (v < w) || (v == w && i < j);
}

// Insert (v,i) into ascending sorted 4-list; stream version (plain <, valid
// because candidate indices are strictly increasing per lane).
__device__ __forceinline__ void top4_insert_fast(float v, int i,
    float& t0, int& i0, float& t1, int& i1,
    float& t2, int& i2, float& t3, int& i3) {
  if (v < t3) {
    t3 = v; i3 = i;
    if (t3 < t2) {
      float tv = t2; int ti = i2; t2 = t3; i2 = i3; t3 = tv; i3 = ti;
      if (t2 < t1) {
        tv = t1; ti = i1; t1 = t2; i1 = i2; t2 = tv; i2 = ti;
        if (t1 < t0) {
          tv = t0; ti = i0; t0 = t1; i0 = i1; t1 = tv; i1 = ti;
        }
      }
    }
  }
}

// Merge version with full tie-breaking.
__device__ __forceinline__ void top4_insert(float v, int i,
    float& t0, int& i0, float& t1, int& i1,
    float& t2, int& i2, float& t3, int& i3) {
  if (pless(v, i, t3, i3)) {
    t3 = v; i3 = i;
    if (pless(t3, i3, t2, i2)) {
      float tv = t2; int ti = i2; t2 = t3; i2 = i3; t3 = tv; i3 = ti;
      if (pless(t2, i2, t1, i1)) {
        tv = t1; ti = i1; t1 = t2; i1 = i2; t2 = tv; i2 = ti;
        if (pless(t1, i1, t0, i0)) {
          tv = t0; ti = i0; t0 = t1; i0 = i1; t1 = tv; i1 = ti;
        }
      }
    }
  }
}

// ---------------------------------------------------------------------------
// Pass 0: pack points as (p0, p1, p2, |p|^2) float4 so the streamed WMMA
// operand is a single ds_load_b64 per lane with zero per-tile VALU prep.
// ---------------------------------------------------------------------------
__global__ __launch_bounds__(256) void pack_points(
    const float* __restrict__ X, const float* __restrict__ Y,
    v4f* __restrict__ packX, v4f* __restrict__ packY) {
  const int i = blockIdx.x * 256 + threadIdx.x;
  if (i < NPTS) {
    const float a0 = X[i * 3 + 0], a1 = X[i * 3 + 1], a2 = X[i * 3 + 2];
    v4f px = {a0, a1, a2, a0 * a0 + a1 * a1 + a2 * a2};
    packX[i] = px;
    const float b0 = Y[i * 3 + 0], b1 = Y[i * 3 + 1], b2 = Y[i * 3 + 2];
    v4f py = {b0, b1, b2, b0 * b0 + b1 * b1 + b2 * b2};
    packY[i] = py;
  }
}

// ---------------------------------------------------------------------------
// Pass 1: per-row min of Dcd.  Block = 4 waves share one 16-row stripe of X;
// packY (128KB) is TDM-staged into LDS; each wave walks every 4th column tile
// with per-(reg,lane) running mins in registers; ds_swizzle row reduction;
// LDS combine across waves.  C layout: (reg r, lane) = (row r+8*(lane>=16),
// col lane&15).
// ---------------------------------------------------------------------------
__global__ __launch_bounds__(128) void rowmin_pass(
    const float* __restrict__ X, const float* __restrict__ packY,
    float* __restrict__ rowmin) {
  __shared__ float sh[4][16];
  const int lane = threadIdx.x & 31;
  const int wv   = threadIdx.x >> 5;
  const int r16  = lane & 15;
  const bool hi  = lane >= 16;
  const int rowBase = blockIdx.x * 16;

  __builtin_prefetch(packY, 0, 1);
  if (wv == 0) tdm_stage(packY, g_lds);

  const float x0 = X[(rowBase + r16) * 3 + 0];
  const float x1 = X[(rowBase + r16) * 3 + 1];
  const float x2 = X[(rowBase + r16) * 3 + 2];
  const float xsq = x0 * x0 + x1 * x1 + x2 * x2;
  // A 16x4: lanes 0-15 hold K0,K1; lanes 16-31 hold K2,K3 (row = lane&15).
  // Fold the -2 here; K=3 slot = 1 picks up ysq from the streamed operand.
  v2f a;
  a.x = hi ? -2.0f * x2 : -2.0f * x0;
  a.y = hi ? 1.0f : -2.0f * x1;

  float rmin[8];
#pragma unroll
  for (int r = 0; r < 8; ++r) rmin[r] = FBIG;

  __syncthreads();           // LDS staging visible to all waves

  const v2f* L2 = (const v2f*)g_lds;
  int pidx = (wv << 5) + ((r16 << 1) | (hi ? 1 : 0));
  for (int tt = 0; tt < NTILE / 4; ++tt) {   // uniform trip count
    const v2f bop = L2[pidx];                // (y0,y1) or (y2,ysq)
    pidx += 128;
    v8f c = {};
    c = wmma_f32_16x16x4(a, bop, c);         // c[r] = ysq[col] - 2*x.y
#pragma unroll
    for (int r = 0; r < 8; ++r) rmin[r] = fmin_fast(rmin[r], c[r]);
  }

#pragma unroll
  for (int r = 0; r < 8; ++r) {
    float m = rmin[r];      // reduce over the 16 columns held by this half
    m = fmin_fast(m, swz_xor_f<1>(m));
    m = fmin_fast(m, swz_xor_f<2>(m));
    m = fmin_fast(m, swz_xor_f<4>(m));
    m = fmin_fast(m, swz_xor_f<8>(m));
    if (r16 == r) sh[wv][r + (hi ? 8 : 0)] = m;   // row-in-tile = r + 8*hi
  }
  __syncthreads();
  if (threadIdx.x < 16) {
    const float m = fmin_fast(fmin_fast(sh[0][lane], sh[1][lane]),
                              fmin_fast(sh[2][lane], sh[3][lane]));
    rowmin[rowBase + lane] = m + xsq;   // lane<16 holds xsq of row 'lane'
  }
}

// ---------------------------------------------------------------------------
// Pass 2: per-column min + argmin of Dcd (-> Dc, indc) and the XX = X[indc]
// gather (served by the packed LDS copy).  Per lane all 8 C-registers belong
// to ONE column, so the running (min, argmin) is one register pair per lane.
// ---------------------------------------------------------------------------
__global__ __launch_bounds__(128) void colmin_pass(
    const float* __restrict__ packX, const float* __restrict__ Y,
    float* __restrict__ colmin, int* __restrict__ indc,
    float* __restrict__ XX, float* __restrict__ xx2) {
  __shared__ float shv[4][16];
  __shared__ int   shi[4][16];
  const int lane = threadIdx.x & 31;
  const int wv   = threadIdx.x >> 5;
  const int r16  = lane & 15;
  const bool hi  = lane >= 16;
  const int colBase = blockIdx.x * 16;

  __builtin_prefetch(packX, 0, 1);
  if (wv == 0) tdm_stage(packX, g_lds);

  const float y0 = Y[(colBase + r16) * 3 + 0];
  const float y1 = Y[(colBase + r16) * 3 + 1];
  const float y2 = Y[(colBase + r16) * 3 + 2];
  const float ysq = y0 * y0 + y1 * y1 + y2 * y2;
  v2f b;                          // -2 folded into the fixed B operand
  b.x = hi ? -2.0f * y2 : -2.0f * y0;
  b.y = hi ? 1.0f : -2.0f * y1;   // K=3 slot = 1 picks up xsq from A's K=3

  float best = FBIG;
  int bidx = 0x7fffffff;

  __syncthreads();

  const v2f* L2 = (const v2f*)g_lds;
  int pidx = (wv << 5) + ((r16 << 1) | (hi ? 1 : 0));
  int base = (wv << 4) + (hi ? 8 : 0);       // candidate row of c[0]
  for (int tt = 0; tt < NTILE / 4; ++tt) {
    const v2f aop = L2[pidx];                // (x0,x1) or (x2,xsq)
    pidx += 128;
    v8f c = {};
    c = wmma_f32_16x16x4(aop, b, c);         // c[r] = xsq[row] - 2*x.y
#pragma unroll
    for (int r = 0; r < 8; ++r) {            // ascending idx -> plain <
      if (c[r] < best) { best = c[r]; bidx = base + r; }
    }
    base += 64;
  }
  // merge the two halves (same column, disjoint row subsets)
  {
    const float ov = swz_xor_f<16>(best);
    const int   oi = swz_xor_i<16>(bidx);
    if (pless(ov, oi, best, bidx)) { best = ov; bidx = oi; }
  }
  if (!hi) { shv[wv][r16] = best; shi[wv][r16] = bidx; }
  __syncthreads();
  if (threadIdx.x < 16) {
    float bv = shv[0][lane]; int bi = shi[0][lane];
#pragma unroll
    for (int w = 1; w < 4; ++w)
      if (pless(shv[w][lane], shi[w][lane], bv, bi)) {
        bv = shv[w][lane]; bi = shi[w][lane];
      }
    const int m = colBase + lane;
    colmin[m] = bv + ysq;
    indc[m] = bi;
    const v4f g = ((const v4f*)g_lds)[bi];   // X[indc[m]] from packed LDS
    XX[m * 3 + 0] = g.x; XX[m * 3 + 1] = g.y; XX[m * 3 + 2] = g.z;
    xx2[m] = g.w;
  }
}

__device__ __forceinline__ float knn_term(
    const float* __restrict__ XX, const float* __restrict__ xx2, int i,
    float xxj0, float xxj1, float xxj2, float xj2, float dy) {
  const float dx = xx2[i] + xj2 -
      2.0f * (XX[i * 3 + 0] * xxj0 + XX[i * 3 + 1] * xxj1 + XX[i * 3 + 2] * xxj2);
  const float d = dx - dy;
  return d * d;
}

// ---------------------------------------------------------------------------
// Pass 3: per-column top-4 of Dyy (k=0 is self, excluded by using k=1..3),
// then Dk[j] = sum_{k=1..3} (Dx[i_k,j] - Dyy[i_k,j])^2, per-block partials.
// ---------------------------------------------------------------------------
__global__ __launch_bounds__(128) void knn_pass(
    const float* __restrict__ packY, const float* __restrict__ Y,
    const float* __restrict__ XX, const float* __restrict__ xx2,
    float* __restrict__ knnPartial) {
  __shared__ float shv[4][16][4];
  __shared__ int   shi[4][16][4];
  __shared__ float shsum[16];
  const int lane = threadIdx.x & 31;
  const int wv   = threadIdx.x >> 5;
  const int r16  = lane & 15;
  const bool hi  = lane >= 16;
  const int colBase = blockIdx.x * 16;

  __builtin_prefetch(packY, 0, 1);
  if (wv == 0) tdm_stage(packY, g_lds);

  const float y0 = Y[(colBase + r16) * 3 + 0];
  const float y1 = Y[(colBase + r16) * 3 + 1];
  const float y2 = Y[(colBase + r16) * 3 + 2];
  const float ysq = y0 * y0 + y1 * y1 + y2 * y2;
  v2f b;
  b.x = hi ? -2.0f * y2 : -2.0f * y0;
  b.y = hi ? 1.0f : -2.0f * y1;

  float t0 = FBIG, t1 = FBIG, t2 = FBIG, t3 = FBIG;
  int i0 = 0x7fffffff, i1 = 0x7fffffff, i2 = 0x7fffffff, i3 = 0x7fffffff;

  __syncthreads();

  const v2f* L2 = (const v2f*)g_lds;
  int pidx = (wv << 5) + ((r16 << 1) | (hi ? 1 : 0));
  int base = (wv << 4) + (hi ? 8 : 0);
  for (int tt = 0; tt < NTILE / 4; ++tt) {
    const v2f aop = L2[pidx];
    pidx += 128;
    v8f c = {};
    c = wmma_f32_16x16x4(aop, b, c);    // c[r] = ysq[row] - 2*y_i.y_j
#pragma unroll
    for (int r = 0; r < 8; ++r)
      top4_insert_fast(c[r], base + r, t0, i0, t1, i1, t2, i2, t3, i3);
    base += 64;
  }
  // merge halves: snapshot other half's sorted list first, then insert
  {
    const float o0 = swz_xor_f<16>(t0);
    const float o1 = swz_xor_f<16>(t1);
    const float o2 = swz_xor_f<16>(t2);
    const float o3 = swz_xor_f<16>(t3);
    const int oi0 = swz_xor_i<16>(i0);
    const int oi1 = swz_xor_i<16>(i1);
    const int oi2 = swz_xor_i<16>(i2);
    const int oi3 = swz_xor_i<16>(i3);
    top4_insert(o0, oi0, t0, i0, t1, i1, t2, i2, t3, i3);
    top4_insert(o1, oi1, t0, i0, t1, i1, t2, i2, t3, i3);
    top4_insert(o2, oi2, t0, i0, t1, i1, t2, i2, t3, i3);
    top4_insert(o3, oi3, t0, i0, t1, i1, t2, i2, t3, i3);
  }
  if (!hi) {
    shv[wv][r16][0] = t0; shv[wv][r16][1] = t1;
    shv[wv][r16][2] = t2; shv[wv][r16][3] = t3;
    shi[wv][r16][0] = i0; shi[wv][r16][1] = i1;
    shi[wv][r16][2] = i2; shi[wv][r16][3] = i3;
  }
  __syncthreads();
  if (threadIdx.x < 16) {
    float u0 = shv[0][lane][0], u1 = shv[0][lane][1];
    float u2 = shv[0][lane][2], u3 = shv[0][lane][3];
    int j0 = shi[0][lane][0], j1 = shi[0][lane][1];
    int j2 = shi[0][lane][2], j3 = shi[0][lane][3];
#pragma unroll
    for (int w = 1; w < 4; ++w)
#pragma unroll
      for (int k = 0; k < 4; ++k)
        top4_insert(shv[w][lane][k], shi[w][lane][k],
                    u0, j0, u1, j1, u2, j2, u3, j3);
    const int j = colBase + lane;
    const float xxj0 = XX[j * 3 + 0];
    const float xxj1 = XX[j * 3 + 1];
    const float xxj2 = XX[j * 3 + 2];
    const float xj2  = xx2[j];
    float acc = 0.0f;
    // k=0 is the nearest neighbor (self); reference sums k=1..3 only.
    acc += knn_term(XX, xx2, j1, xxj0, xxj1, xxj2, xj2, u1 + ysq);
    acc += knn_term(XX, xx2, j2, xxj0, xxj1, xxj2, xj2, u2 + ysq);
    acc += knn_term(XX, xx2, j3, xxj0, xxj1, xxj2, xj2, u3 + ysq);
    shsum[threadIdx.x] = acc;
    (void)j0;
  }
  __syncthreads();
  if (threadIdx.x == 0) {
    float s = 0.0f;
#pragma unroll
    for (int k = 0; k < 16; ++k) s += shsum[k];
    knnPartial[blockIdx.x] = s;
  }
}

// ---------------------------------------------------------------------------
// Final reduction: d_out[0] = mean(rowmin) + mean(colmin), d_out[1] = Dknn.
// ---------------------------------------------------------------------------
__global__ __launch_bounds__(256) void final_reduce(
    const float* __restrict__ rowmin, const float* __restrict__ colmin,
    const float* __restrict__ knnPartial, float* __restrict__ out) {
  __shared__ float sA[256], sB[256], sC[256];
  float a = 0.0f, b = 0.0f, c = 0.0f;
  for (int i = threadIdx.x; i < NPTS; i += 256) { a += rowmin[i]; b += colmin[i]; }
  for (int i = threadIdx.x; i < NTILE; i += 256) c += knnPartial[i];
  sA[threadIdx.x] = a; sB[threadIdx.x] = b; sC[threadIdx.x] = c;
  __syncthreads();
  for (int s = 128; s > 0; s >>= 1) {
    if ((int)threadIdx.x < s) {
      sA[threadIdx.x] += sA[threadIdx.x + s];
      sB[threadIdx.x] += sB[threadIdx.x + s];
      sC[threadIdx.x] += sC[threadIdx.x + s];
    }
    __syncthreads();
  }
  if (threadIdx.x == 0) {
    out[0] = sA[0] * (1.0f / NPTS) + sB[0] * (1.0f / NPTS);  // d_ch
    out[1] = sC[0];                                          // Dknn
  }
}

extern "C" void kernel_launch(void* const* d_in, const int* in_sizes, int n_in,
                              void* d_out, int out_size, void* d_ws, size_t ws_size,
                              hipStream_t stream) {
  (void)in_sizes; (void)n_in; (void)out_size; (void)ws_size;
  const float* X = (const float*)d_in[0];   // [1,8192,3] f32
  const float* Y = (const float*)d_in[1];   // [1,8192,3] f32
  // d_in[2], d_in[3] are dead forward args.

  float* ws     = (float*)d_ws;
  float* packX  = ws;                       // 4*NPTS floats (16B aligned)
  float* packY  = ws + 4 * NPTS;            // 4*NPTS floats
  float* rowmin = ws + 8 * NPTS;            // NPTS floats
  float* colmin = ws + 9 * NPTS;            // NPTS floats
  int*   indc   = (int*)(ws + 10 * NPTS);   // NPTS ints
  float* XX     = ws + 11 * NPTS;           // 3*NPTS floats
  float* xx2    = ws + 14 * NPTS;           // NPTS floats
  float* knnP   = ws + 15 * NPTS;           // NTILE floats

  const dim3 blk(128);       // 4 wave32s
  const dim3 grd(NTILE);     // 512 blocks, one 16-wide stripe each
  pack_points<<<dim3(NPTS / 256), dim3(256), 0, stream>>>(
      X, Y, (v4f*)packX, (v4f*)packY);
  rowmin_pass<<<grd, blk, STAGE_BYTES, stream>>>(X, packY, rowmin);
  colmin_pass<<<grd, blk, STAGE_BYTES, stream>>>(packX, Y, colmin, indc, XX, xx2);
  knn_pass<<<grd, blk, STAGE_BYTES, stream>>>(packY, Y, XX, xx2, knnP);
  final_reduce<<<dim3(1), dim3(256), 0, stream>>>(rowmin, colmin, knnP,
                                                  (float*)d_out);
}